// SingleQubitOperationLayerV2_27470610825418
// MI455X (gfx1250) — compile-verified
//
#include <hip/hip_runtime.h>

// ============================================================================
// SingleQubitOperationLayerV2 (n=12, N=4096, B=2048), MI455X / gfx1250.
//
// Identity:  D_phi * BS * D_theta * BS * D_in  ==  G1 (x) G2
//   (BS = kron^12(bs); every diagonal is bit-separable, so the whole chain
//    factorizes over the high-6/low-6 qubit split; M = kron^6(bs) with
//    M[r][c] = (1/8) * i^popcount(r^c), symmetric.)
// Per batch row:  OUT_b(64x64) = G1 @ Z_b @ G2^T  (complex f32).
//
// Roofline: traffic = 64 MiB in + 64 MiB out -> ~5.5us @ 23.3 TB/s HBM.
// Compute = 2 complex 64^3 GEMMs x 2048 rows ~ 8.6 GFLOP -> trivially hidden
// by f32 WMMA (v_wmma_f32_16x16x4_f32), keeping full f32 precision.
//
// G1/G2^T staged to LDS once per block, 4 batch rows per block; Z tile loaded
// via the Tensor Data Mover with hardware LDS row padding (pad 4 DW every
// 64 DW == 68-float stride), synced with s_wait_tensorcnt + barrier.
// k-loops fully unrolled so LDS accesses become immediate-offset ds_loads.
// ============================================================================

typedef __attribute__((ext_vector_type(2))) float v2f;
typedef __attribute__((ext_vector_type(8))) float v8f;

#define NN     4096
#define DIM    64
#define LSTR   68              // padded LDS row stride (floats)
#define PL     (DIM * LSTR)    // one 64x64 padded plane
#define NBATCH 4

#if __has_builtin(__builtin_amdgcn_tensor_load_to_lds) && \
    __has_builtin(__builtin_amdgcn_s_wait_tensorcnt)
#define USE_TDM 1
#else
#define USE_TDM 0
#endif

__device__ __forceinline__ v8f wmma_f32(v2f a, v2f b, v8f c) {
  // D = A(16x4) * B(4x16) + C(16x16), all f32. (No A/B neg for f32 per ISA.)
  return __builtin_amdgcn_wmma_f32_16x16x4_f32(
      false, a, false, b, (short)0, c, false, false);
}

#if USE_TDM
typedef __attribute__((ext_vector_type(4))) unsigned int v4u;
typedef __attribute__((ext_vector_type(8))) int v8i;
typedef __attribute__((ext_vector_type(4))) int v4i;

// Generic LDS pointer -> LDS byte offset: the LDS aperture lives entirely in
// addr[63:32] (ISA 10.2), so addr[31:0] is the in-LDS byte offset.
__device__ __forceinline__ unsigned lds_byte_off(const void* p) {
  return (unsigned)(uintptr_t)p;
}

// DMA one contiguous 4096-float plane (16 KiB) from global into LDS, with
// hardware padding: +4 DWORDs after every 64 DWORDs -> 68-float row stride.
// D# per ISA ch.8: group0 = {count, lds_addr, global_addr, type=2},
// group1 = {mask=0, data_size=4B, pad_enable, pad_interval=64dw,
//           pad_amount=4dw, tensor_dim0=tile_dim0=4096, dims1..=1/0}.
__device__ __forceinline__ void tdm_load_plane(const float* gsrc,
                                               unsigned lds_off) {
  const unsigned long long ga = (unsigned long long)(uintptr_t)gsrc;
  v4u g0;
  g0.x = 1u;                                  // count=1, user desc
  g0.y = lds_off;                             // LDS byte address
  g0.z = (unsigned)ga;                        // global addr [31:0]
  g0.w = (unsigned)(ga >> 32) | (2u << 30);   // addr[56:32] | type=2 (image)

  v8i g1;
  g1[0] = (int)((2u << 16)      // data_size = 4 bytes
              | (1u << 20)      // pad_enable
              | (5u << 22)      // pad_interval: 2^(5+1) = 64 DWORDs
              | (3u << 25));    // pad_amount: 3+1 = 4 DWORDs
  g1[1] = (int)(4096u << 16);   // tensor_dim0[15:0] @ bits63:48
  g1[2] = (int)(1u << 16);      // tensor_dim0 hi = 0 ; tensor_dim1 = 1
  g1[3] = (int)(4096u << 16);   // tensor_dim1 hi = 0 ; tile_dim0 = 4096
  g1[4] = 0;                    // tile_dim1 = 0 (unused), tile_dim2 = 0
  g1[5] = 4096;                 // tensor_dim0_stride lo
  g1[6] = 0;                    // stride0 hi, stride1 lo
  g1[7] = 0;

  v4i gz; gz[0] = 0; gz[1] = 0; gz[2] = 0; gz[3] = 0;
#if __clang_major__ >= 23
  v8i gz8; for (int i = 0; i < 8; ++i) gz8[i] = 0;
  __builtin_amdgcn_tensor_load_to_lds(g0, g1, gz, gz, gz8, 0);
#else
  __builtin_amdgcn_tensor_load_to_lds(g0, g1, gz, gz, 0);
#endif
}
#endif  // USE_TDM

// ---------------------------------------------------------------------------
// Build G1 (side 0) and G2^T (side 1): G = D_p * M * D_t * M * D_s.
// Workspace layout: [G1re | G1im | G2Tre | G2Tim], each 64*64 floats.
// ---------------------------------------------------------------------------
__global__ void build_factors_kernel(const float* __restrict__ alphas,
                                     const float* __restrict__ betas,
                                     const float* __restrict__ thetas,
                                     const float* __restrict__ phis,
                                     float* __restrict__ G) {
  const int s = blockIdx.x;   // 0: high-6 qubits (G1), 1: low-6 qubits (G2^T)
  const int j = threadIdx.x;  // 0..63, owns column j

  __shared__ float sre[DIM], sim_[DIM];
  __shared__ float tre[DIM], tim[DIM];
  __shared__ float pre_[DIM], pim_[DIM];
  __shared__ float Tre[DIM][DIM], Tim[DIM][DIM];

  float a_in = 0.f, a_t = 0.f, a_p = 0.f;
  for (int q = 0; q < 6; ++q) {
    const int bit = (j >> (5 - q)) & 1;
    const int qg = s * 6 + q;
    a_in += bit ? betas[qg] : alphas[qg];
    if (!bit) { a_t += thetas[qg]; a_p += phis[qg]; }
  }
  sre[j]  = cosf(a_in); sim_[j] = sinf(a_in);
  tre[j]  = cosf(a_t);  tim[j]  = sinf(a_t);
  pre_[j] = cosf(a_p);  pim_[j] = sinf(a_p);
  __syncthreads();

  const float sjre = sre[j], sjim = sim_[j];

  // T[:,j] = D_t * M * D_s (column j)
  for (int i = 0; i < DIM; ++i) {
    const int k4 = __popc(i ^ j) & 3;
    const float mre = (k4 == 0) ? 0.125f : ((k4 == 2) ? -0.125f : 0.f);
    const float mim = (k4 == 1) ? 0.125f : ((k4 == 3) ? -0.125f : 0.f);
    const float msre = mre * sjre - mim * sjim;
    const float msim = mre * sjim + mim * sjre;
    Tre[i][j] = tre[i] * msre - tim[i] * msim;
    Tim[i][j] = tre[i] * msim + tim[i] * msre;
  }
  __syncthreads();

  float* Gre = G + (s ? 2 * NN : 0);
  float* Gim = Gre + NN;

  for (int i = 0; i < DIM; ++i) {
    float ar = 0.f, ai = 0.f;
    for (int k = 0; k < DIM; ++k) {
      const int k4 = __popc(i ^ k) & 3;
      const float mre = (k4 == 0) ? 0.125f : ((k4 == 2) ? -0.125f : 0.f);
      const float mim = (k4 == 1) ? 0.125f : ((k4 == 3) ? -0.125f : 0.f);
      ar += mre * Tre[k][j] - mim * Tim[k][j];
      ai += mre * Tim[k][j] + mim * Tre[k][j];
    }
    const float gre = pre_[i] * ar - pim_[i] * ai;
    const float gim = pre_[i] * ai + pim_[i] * ar;
    if (s == 0) { Gre[i * DIM + j] = gre; Gim[i * DIM + j] = gim; }
    else        { Gre[j * DIM + i] = gre; Gim[j * DIM + i] = gim; }
  }
}

// ---------------------------------------------------------------------------
// Main kernel: per batch row OUT = G1 @ Z @ G2^T via f32 WMMA.
// 512 blocks x 256 threads (8 wave32); each block does 4 batch rows,
// reusing LDS-resident G1/G2^T. Fragment lane map (ISA 7.12.2, f32):
//   lane = kh*16 + m;  A: {A[m][4kk+2kh], A[m][4kk+2kh+1]}
//                      B: {B[4kk+2kh][n], B[4kk+2kh+1][n]}
//                      C/D vgpr v: C[v + 8*kh][n]
// ---------------------------------------------------------------------------
__global__ void __launch_bounds__(256)
qkron_apply_kernel(const float* __restrict__ x,
                   const float* __restrict__ G,
                   float* __restrict__ out) {
  __shared__ __align__(16) float smem[8 * PL];
  float* Zre  = smem + 0 * PL;
  float* Zim  = smem + 1 * PL;
  float* Wre  = smem + 2 * PL;
  float* Wim  = smem + 3 * PL;
  float* g1re = smem + 4 * PL;
  float* g1im = smem + 5 * PL;
  float* g2re = smem + 6 * PL;   // G2^T
  float* g2im = smem + 7 * PL;

  const int t = threadIdx.x;

  // Stage the four 64x64 G planes into padded LDS once per block.
#pragma unroll
  for (int p = 0; p < 4; ++p) {
    const float* src = G + p * NN;
    float* dst = smem + (4 + p) * PL;
#pragma unroll
    for (int k = 0; k < 4; ++k) {
      const int c = t + k * 256;          // float4 chunk id, 16 per row
      const int row = c >> 4;
      const int col = (c & 15) * 4;
      *(float4*)(&dst[row * LSTR + col]) = ((const float4*)src)[c];
    }
  }

  const int wave = t >> 5;
  const int lane = t & 31;
  const int m16  = lane & 15;   // M (A rows) / N (B,C cols) within a tile
  const int kh   = lane >> 4;   // K half-select

  for (int ib = 0; ib < NBATCH; ++ib) {
    const int b = blockIdx.x * NBATCH + ib;
    const float* xr = x + (size_t)b * (2 * NN);
    const float* xi = xr + NN;

#if USE_TDM
    // One wave issues two TDM DMAs (re/im planes) with HW row padding.
    if (t < 32) {
      tdm_load_plane(xr, lds_byte_off(Zre));
      tdm_load_plane(xi, lds_byte_off(Zim));
      __builtin_amdgcn_s_wait_tensorcnt(0);
    }
#else
#pragma unroll
    for (int k = 0; k < 4; ++k) {
      const int c = t + k * 256;
      const int row = c >> 4;
      const int col = (c & 15) * 4;
      *(float4*)(&Zre[row * LSTR + col]) = ((const float4*)xr)[c];
      *(float4*)(&Zim[row * LSTR + col]) = ((const float4*)xi)[c];
    }
#endif
    __syncthreads();   // Z (and on ib==0, G) visible to all waves

    // ---------------- GEMM 1: W = G1 @ Z ----------------
#pragma unroll
    for (int half = 0; half < 2; ++half) {
      const int tile = wave + half * 8;
      const int tr = tile >> 2, tc = tile & 3;
      const int arow = tr * 16 + m16;
      const int bcol = tc * 16 + m16;
      v8f cre = {};
      v8f cim = {};
#pragma unroll
      for (int kk = 0; kk < 16; ++kk) {
        const int ka = kk * 4 + kh * 2;
        v2f are, aim, naim, bre, bim;
        are.x = g1re[arow * LSTR + ka];  are.y = g1re[arow * LSTR + ka + 1];
        aim.x = g1im[arow * LSTR + ka];  aim.y = g1im[arow * LSTR + ka + 1];
        bre.x = Zre[(ka + 0) * LSTR + bcol];  bre.y = Zre[(ka + 1) * LSTR + bcol];
        bim.x = Zim[(ka + 0) * LSTR + bcol];  bim.y = Zim[(ka + 1) * LSTR + bcol];
        naim.x = -aim.x; naim.y = -aim.y;
        cre = wmma_f32(are,  bre, cre);   // += Are*Bre
        cre = wmma_f32(naim, bim, cre);   // -= Aim*Bim
        cim = wmma_f32(are,  bim, cim);   // += Are*Bim
        cim = wmma_f32(aim,  bre, cim);   // += Aim*Bre
      }
#pragma unroll
      for (int v = 0; v < 8; ++v) {
        const int r = tr * 16 + v + 8 * kh;
        const int c = tc * 16 + m16;
        Wre[r * LSTR + c] = cre[v];
        Wim[r * LSTR + c] = cim[v];
      }
    }
    __syncthreads();   // all Z reads done, W published

    // ---------------- GEMM 2: OUT = W @ G2^T ----------------
    float* outr = out + (size_t)b * (2 * NN);
    float* outi = outr + NN;
#pragma unroll
    for (int half = 0; half < 2; ++half) {
      const int tile = wave + half * 8;
      const int tr = tile >> 2, tc = tile & 3;
      const int arow = tr * 16 + m16;
      const int bcol = tc * 16 + m16;
      v8f cre = {};
      v8f cim = {};
#pragma unroll
      for (int kk = 0; kk < 16; ++kk) {
        const int ka = kk * 4 + kh * 2;
        v2f are, aim, naim, bre, bim;
        are.x = Wre[arow * LSTR + ka];  are.y = Wre[arow * LSTR + ka + 1];
        aim.x = Wim[arow * LSTR + ka];  aim.y = Wim[arow * LSTR + ka + 1];
        bre.x = g2re[(ka + 0) * LSTR + bcol];  bre.y = g2re[(ka + 1) * LSTR + bcol];
        bim.x = g2im[(ka + 0) * LSTR + bcol];  bim.y = g2im[(ka + 1) * LSTR + bcol];
        naim.x = -aim.x; naim.y = -aim.y;
        cre = wmma_f32(are,  bre, cre);
        cre = wmma_f32(naim, bim, cre);
        cim = wmma_f32(are,  bim, cim);
        cim = wmma_f32(aim,  bre, cim);
      }
#pragma unroll
      for (int v = 0; v < 8; ++v) {
        const int r = tr * 16 + v + 8 * kh;
        const int c = tc * 16 + m16;
        outr[r * DIM + c] = cre[v];
        outi[r * DIM + c] = cim[v];
      }
    }
    // No extra barrier needed: next iteration only overwrites Z, and the
    // GEMM1->GEMM2 barrier already guarantees all Z reads completed.
  }
}

// ---------------------------------------------------------------------------
extern "C" void kernel_launch(void* const* d_in, const int* in_sizes, int n_in,
                              void* d_out, int out_size, void* d_ws, size_t ws_size,
                              hipStream_t stream) {
  const float* x      = (const float*)d_in[0];
  const float* alphas = (const float*)d_in[1];
  const float* betas  = (const float*)d_in[2];
  const float* thetas = (const float*)d_in[3];
  const float* phis   = (const float*)d_in[4];
  float* G = (float*)d_ws;   // 4 * 4096 floats = 64 KiB scratch

  build_factors_kernel<<<2, 64, 0, stream>>>(alphas, betas, thetas, phis, G);
  qkron_apply_kernel<<<2048 / NBATCH, 256, 0, stream>>>(x, G, (float*)d_out);
}